// GFN_AE_15522011808115
// MI455X (gfx1250) — compile-verified
//
#include <hip/hip_runtime.h>
#include <stdint.h>

// Problem dimensions (from the reference: S, N, L, B = 200000, 250000, 64, 64)
#define S_DIM 200000
#define N_DIM 250000
#define L_DIM 64
#define B_DIM 64

typedef float v2f __attribute__((ext_vector_type(2)));
typedef float v8f __attribute__((ext_vector_type(8)));

// ---------------------------------------------------------------------------
// Utility: zero a u32 region (grid-stride)
// ---------------------------------------------------------------------------
__global__ void k_zero(uint32_t* __restrict__ p, int n) {
    int i = blockIdx.x * blockDim.x + threadIdx.x;
    int stride = gridDim.x * blockDim.x;
    for (; i < n; i += stride) p[i] = 0u;
}

// ---------------------------------------------------------------------------
// K1: added[n] = (nn_m[nn_n[n]] != n); c_m[m] = #added n with nn_n[n]==m
// ---------------------------------------------------------------------------
__global__ void k_added(const int* __restrict__ nn_n, const int* __restrict__ nn_m,
                        int* __restrict__ added, int* __restrict__ c_m) {
    int n = blockIdx.x * blockDim.x + threadIdx.x;
    if (n >= N_DIM) return;
    int m = nn_n[n];
    int a = (nn_m[m] != n) ? 1 : 0;
    added[n] = a;
    if (a) atomicAdd(&c_m[m], 1);
}

// ---------------------------------------------------------------------------
// K2: scatter scaled We rows: We_n[nn_m[m]] += We_m[m] / (c_m[m]+1)
//     also cnt_n[nn_m[m]] += 1
// ---------------------------------------------------------------------------
__global__ void k_scatter_We(const float* __restrict__ We_m, const int* __restrict__ nn_m,
                             const int* __restrict__ c_m, float* __restrict__ We_n,
                             int* __restrict__ cnt_n) {
    int e = blockIdx.x * blockDim.x + threadIdx.x;  // S*64 = 12.8M < 2^31
    if (e >= S_DIM * 64) return;
    int m = e >> 6;
    int l = e & 63;
    float scale = 1.0f / (float)(c_m[m] + 1);
    int n = nn_m[m];
    unsafeAtomicAdd(&We_n[n * 64 + l], We_m[e] * scale);
    if (l == 0) atomicAdd(&cnt_n[n], 1);
}

// ---------------------------------------------------------------------------
// K3: added-node contributions: We_n[n] += We_m[nn_n[n]] / (c_m[nn_n[n]]+1)
//     (runs after K2; each row n touched exactly once -> plain RMW)
// ---------------------------------------------------------------------------
__global__ void k_add_added_We(const float* __restrict__ We_m, const int* __restrict__ nn_n,
                               const int* __restrict__ c_m, const int* __restrict__ added,
                               float* __restrict__ We_n) {
    int e = blockIdx.x * blockDim.x + threadIdx.x;  // N*64 = 16M
    if (e >= N_DIM * 64) return;
    int n = e >> 6;
    if (!added[n]) return;
    int l = e & 63;
    int m = nn_n[n];
    float scale = 1.0f / (float)(c_m[m] + 1);
    We_n[e] += We_m[m * 64 + l] * scale;
}

// ---------------------------------------------------------------------------
// K4: encode matmul  z[64,64] += x[64,N] @ We_n[N,64]  (K split over blocks)
//     fp32 WMMA 16x16x4. 8 waves/block, each wave owns two 16x16 z tiles.
// ---------------------------------------------------------------------------
__global__ __launch_bounds__(256) void k_encode(const float* __restrict__ x,
                                                const float* __restrict__ We_n,
                                                float* __restrict__ zbuf) {
    const int KC = 1024;  // K chunk per block
    const int KS = 32;    // K per LDS stage
    __shared__ float xs[64][KS + 1];   // xs[b][k]
    __shared__ float ws[KS][64 + 1];   // ws[k][l]

    int t = threadIdx.x;
    int wave = t >> 5, lane = t & 31;
    int laneLo = lane & 15, laneHi = lane >> 4;
    int mi = wave & 3;          // z tile row   (b-dim)
    int nb = wave >> 2;         // tile-col pair (l-dim): cols 2*nb, 2*nb+1
    int ni0 = 2 * nb, ni1 = 2 * nb + 1;

    long long k0base = (long long)blockIdx.x * KC;
    v8f acc0 = {}; v8f acc1 = {};

    for (int stage = 0; stage < KC / KS; ++stage) {
        long long k0 = k0base + (long long)stage * KS;
        // stage x tile: 64 rows (b) x KS cols (k); 32 threads per row, coalesced
        for (int i = 0; i < (64 * KS) / 256; ++i) {
            int e = t + 256 * i;
            int b = e / KS, kk = e % KS;
            long long k = k0 + kk;
            xs[b][kk] = (k < N_DIM) ? x[(long long)b * N_DIM + k] : 0.0f;
        }
        // stage We_n tile: KS rows (k) x 64 cols (l); coalesced 256B rows
        for (int i = 0; i < (KS * 64) / 256; ++i) {
            int e = t + 256 * i;
            int kk = e >> 6, l = e & 63;
            long long k = k0 + kk;
            ws[kk][l] = (k < N_DIM) ? We_n[k * 64 + l] : 0.0f;
        }
        __syncthreads();
        for (int ks = 0; ks < KS / 4; ++ks) {
            // A (16x4 f32): vgpr j, lane -> A[m=laneLo, k=2*laneHi+j]
            v2f a;
            a[0] = xs[mi * 16 + laneLo][ks * 4 + 2 * laneHi + 0];
            a[1] = xs[mi * 16 + laneLo][ks * 4 + 2 * laneHi + 1];
            // B (4x16 f32): vgpr j, lane -> B[k=2*laneHi+j, n=laneLo]
            v2f b0, b1;
            b0[0] = ws[ks * 4 + 2 * laneHi + 0][ni0 * 16 + laneLo];
            b0[1] = ws[ks * 4 + 2 * laneHi + 1][ni0 * 16 + laneLo];
            b1[0] = ws[ks * 4 + 2 * laneHi + 0][ni1 * 16 + laneLo];
            b1[1] = ws[ks * 4 + 2 * laneHi + 1][ni1 * 16 + laneLo];
            acc0 = __builtin_amdgcn_wmma_f32_16x16x4_f32(false, a, false, b0,
                                                         (short)0, acc0, false, false);
            acc1 = __builtin_amdgcn_wmma_f32_16x16x4_f32(false, a, false, b1,
                                                         (short)0, acc1, false, false);
        }
        __syncthreads();
    }
    // accumulate partial z into global (C/D layout: vgpr j -> M=j (+8 for hi lanes))
    for (int j = 0; j < 8; ++j) {
        int mrow = mi * 16 + j + 8 * laneHi;
        unsafeAtomicAdd(&zbuf[mrow * 64 + ni0 * 16 + laneLo], acc0[j]);
        unsafeAtomicAdd(&zbuf[mrow * 64 + ni1 * 16 + laneLo], acc1[j]);
    }
}

// ---------------------------------------------------------------------------
// K5: decode matmul  ybT[S,64]: ybT[m,b] = sum_l Wd[l,m]*(z[b,l]+be[l]) + bd[m]
//     M-dim = m (64 per block), N-dim = b (64), K = l (64). fp32 WMMA.
// ---------------------------------------------------------------------------
__global__ __launch_bounds__(256) void k_decode(const float* __restrict__ Wd,
                                                const float* __restrict__ zbuf,
                                                const float* __restrict__ be,
                                                const float* __restrict__ bd,
                                                float* __restrict__ ybT) {
    __shared__ float wds[64][65];  // wds[l][mloc]  (transposed view of Wd tile)
    __shared__ float zs[64][65];   // zs[l][b] = z[b,l] + be[l]
    __shared__ float bds[64];

    int t = threadIdx.x;
    int wave = t >> 5, lane = t & 31;
    int laneLo = lane & 15, laneHi = lane >> 4;
    int mi = wave & 3;            // m tile
    int nb = wave >> 2;           // b tile pair
    int ni0 = 2 * nb, ni1 = 2 * nb + 1;
    int m0 = blockIdx.x * 64;     // S divisible by 64 (3125 blocks)

    for (int i = 0; i < 16; ++i) {
        int e = t + 256 * i;
        int r = e >> 6, c = e & 63;
        wds[r][c] = Wd[r * S_DIM + m0 + c];          // r = l, c = mloc (coalesced)
        zs[c][r] = zbuf[r * 64 + c] + be[c];         // r = b, c = l
    }
    if (t < 64) bds[t] = bd[m0 + t];
    __syncthreads();

    v8f acc0 = {}; v8f acc1 = {};
    for (int ks = 0; ks < 16; ++ks) {
        // A[m, l] = WdT tile
        v2f a;
        a[0] = wds[ks * 4 + 2 * laneHi + 0][mi * 16 + laneLo];
        a[1] = wds[ks * 4 + 2 * laneHi + 1][mi * 16 + laneLo];
        // B[l, b] = zfT tile
        v2f b0, b1;
        b0[0] = zs[ks * 4 + 2 * laneHi + 0][ni0 * 16 + laneLo];
        b0[1] = zs[ks * 4 + 2 * laneHi + 1][ni0 * 16 + laneLo];
        b1[0] = zs[ks * 4 + 2 * laneHi + 0][ni1 * 16 + laneLo];
        b1[1] = zs[ks * 4 + 2 * laneHi + 1][ni1 * 16 + laneLo];
        acc0 = __builtin_amdgcn_wmma_f32_16x16x4_f32(false, a, false, b0,
                                                     (short)0, acc0, false, false);
        acc1 = __builtin_amdgcn_wmma_f32_16x16x4_f32(false, a, false, b1,
                                                     (short)0, acc1, false, false);
    }
    for (int j = 0; j < 8; ++j) {
        int mloc = mi * 16 + j + 8 * laneHi;
        float bias = bds[mloc];
        ybT[(long long)(m0 + mloc) * 64 + ni0 * 16 + laneLo] = acc0[j] + bias;
        ybT[(long long)(m0 + mloc) * 64 + ni1 * 16 + laneLo] = acc1[j] + bias;
    }
}

// ---------------------------------------------------------------------------
// K5b1: segment-sum scatter: outAcc[nn_m[m]] += ybT[m]
// ---------------------------------------------------------------------------
__global__ void k_scatter_out(const float* __restrict__ ybT, const int* __restrict__ nn_m,
                              float* __restrict__ outAcc) {
    int e = blockIdx.x * blockDim.x + threadIdx.x;
    if (e >= S_DIM * 64) return;
    int m = e >> 6;
    int b = e & 63;
    unsafeAtomicAdd(&outAcc[nn_m[m] * 64 + b], ybT[e]);
}

// ---------------------------------------------------------------------------
// K5b2: added contributions: outAcc[n] += ybT[nn_n[n]]  (rows unique -> plain RMW)
// ---------------------------------------------------------------------------
__global__ void k_gather_added_out(const float* __restrict__ ybT, const int* __restrict__ nn_n,
                                   const int* __restrict__ added, float* __restrict__ outAcc) {
    int e = blockIdx.x * blockDim.x + threadIdx.x;
    if (e >= N_DIM * 64) return;
    int n = e >> 6;
    if (!added[n]) return;
    int b = e & 63;
    outAcc[e] += ybT[nn_n[n] * 64 + b];
}

// ---------------------------------------------------------------------------
// K6: finalize: out[b,n] = outAcc[n,b] / max(cnt_n[n]+added[n],1)
//     LDS-tiled transpose so both global read and write are coalesced.
// ---------------------------------------------------------------------------
__global__ __launch_bounds__(256) void k_finalize(const float* __restrict__ outAcc,
                                                  const int* __restrict__ cnt_n,
                                                  const int* __restrict__ added,
                                                  float* __restrict__ out) {
    __shared__ float tile[64][65];
    __shared__ float dens[64];
    int n0 = blockIdx.x * 64;
    int t = threadIdx.x;
    if (t < 64) {
        int n = n0 + t;
        float d = 1.0f;
        if (n < N_DIM) {
            int c = cnt_n[n] + added[n];
            d = 1.0f / (float)(c > 1 ? c : 1);
        }
        dens[t] = d;
    }
    // read phase: coalesced 256B rows of outAcc
    int b = t & 63, irow = t >> 6;
    for (int ii = 0; ii < 16; ++ii) {
        int i = ii * 4 + irow;
        int n = n0 + i;
        tile[i][b] = (n < N_DIM) ? outAcc[(long long)n * 64 + b] : 0.0f;
    }
    __syncthreads();
    // write phase: coalesced along n in the [B,N] output
    int i2 = t & 63, brow = t >> 6;
    int n = n0 + i2;
    if (n < N_DIM) {
        float d = dens[i2];
        for (int bb = 0; bb < 16; ++bb) {
            int b2 = bb * 4 + brow;
            out[(long long)b2 * N_DIM + n] = tile[i2][b2] * d;
        }
    }
}

// ---------------------------------------------------------------------------
// Launcher
// ---------------------------------------------------------------------------
extern "C" void kernel_launch(void* const* d_in, const int* in_sizes, int n_in,
                              void* d_out, int out_size, void* d_ws, size_t ws_size,
                              hipStream_t stream) {
    const float* We_m = (const float*)d_in[0];
    const float* be_m = (const float*)d_in[1];
    const float* Wd_m = (const float*)d_in[2];
    const float* bd_m = (const float*)d_in[3];
    const float* x    = (const float*)d_in[4];
    const int*   nn_n = (const int*)d_in[5];
    const int*   nn_m = (const int*)d_in[6];
    float* out = (float*)d_out;

    // workspace carve (~118 MB). bufA reused: We_n (encode) then outAcc (decode).
    char* w = (char*)d_ws;
    float* bufA = (float*)w;            w += (size_t)N_DIM * 64 * sizeof(float);  // 64 MB
    float* ybT  = (float*)w;            w += (size_t)S_DIM * 64 * sizeof(float);  // 51.2 MB
    float* zbuf = (float*)w;            w += (size_t)64 * 64 * sizeof(float);
    int*   c_m  = (int*)w;              w += (size_t)S_DIM * sizeof(int);
    int*   cnt  = (int*)w;              w += (size_t)N_DIM * sizeof(int);
    int*   added = (int*)w;             w += (size_t)N_DIM * sizeof(int);

    const int T = 256;
    // zero scratch used for accumulation
    k_zero<<<4096, T, 0, stream>>>((uint32_t*)bufA, N_DIM * 64);
    k_zero<<<16,   T, 0, stream>>>((uint32_t*)zbuf, 64 * 64);
    k_zero<<<1024, T, 0, stream>>>((uint32_t*)c_m, S_DIM);
    k_zero<<<1024, T, 0, stream>>>((uint32_t*)cnt, N_DIM);

    // phase 1: added flags + per-master added counts
    k_added<<<(N_DIM + T - 1) / T, T, 0, stream>>>(nn_n, nn_m, added, c_m);

    // phase 2: build We_n (bufA) via scatter, plus segment counts
    k_scatter_We<<<(S_DIM * 64 + T - 1) / T, T, 0, stream>>>(We_m, nn_m, c_m, bufA, cnt);
    k_add_added_We<<<(N_DIM * 64 + T - 1) / T, T, 0, stream>>>(We_m, nn_n, c_m, added, bufA);

    // phase 3: z = x @ We_n (WMMA f32, K split over blocks, atomic accumulate)
    k_encode<<<(N_DIM + 1023) / 1024, T, 0, stream>>>(x, bufA, zbuf);

    // phase 4: ybT = WdT @ (z+be)^T + bd  (WMMA f32)
    k_decode<<<S_DIM / 64, T, 0, stream>>>(Wd_m, zbuf, be_m, bd_m, ybT);

    // phase 5: reuse bufA as outAcc: zero, scatter by nn_m, add "added" gathers
    k_zero<<<4096, T, 0, stream>>>((uint32_t*)bufA, N_DIM * 64);
    k_scatter_out<<<(S_DIM * 64 + T - 1) / T, T, 0, stream>>>(ybT, nn_m, bufA);
    k_gather_added_out<<<(N_DIM * 64 + T - 1) / T, T, 0, stream>>>(ybT, nn_n, added, bufA);

    // phase 6: normalize + transpose into [B, N] output
    k_finalize<<<(N_DIM + 63) / 64, T, 0, stream>>>(bufA, cnt, added, out);
}